// GatedDeltaProduct_13752485282237
// MI455X (gfx1250) — compile-verified
//
#include <hip/hip_runtime.h>
#include <hip/hip_bf16.h>
#include <math.h>

// ---------------- problem constants ----------------
constexpr int Bsz = 2;
constexpr int Tt  = 1024;
constexpr int HID = 1024;
constexpr int Hh  = 4;
constexpr int DK  = 128;
constexpr int DV  = 256;
constexpr int NHn = 2;
constexpr int KSZ = 4;
constexpr float EPSc = 1e-5f;

constexpr int Mrows = Bsz * Tt;          // 2048 rows for all GEMMs

// scan sharding: split DV across NSLICE workgroups per (b,h)
constexpr int NSLICE = 8;
constexpr int SLC = DV / NSLICE;         // 32 columns per block
constexpr int PARTS = 8;                 // threads per column (k-rows split 8 ways)
constexpr int KPT = DK / PARTS;          // 16 state rows per thread

typedef __attribute__((ext_vector_type(2))) float v2f;
typedef __attribute__((ext_vector_type(8))) float v8f;

__device__ __forceinline__ float sigmoidf_(float x) { return 1.f / (1.f + __expf(-x)); }
__device__ __forceinline__ float siluf_(float x)    { return x * sigmoidf_(x); }

// ---------------------------------------------------------------------------
// f32 WMMA GEMM: C[M,N] = A[M,K] @ B[K,N], row-major, M%16==0, N%16==0, K%4==0
// One wave per 16x16 tile, V_WMMA_F32_16X16X4_F32 along K.
// A layout (16x4 f32): v0 = K{0 | 2} split at lane16, v1 = K{1 | 3}.
// B layout (4x16 f32): mirror of A. C/D: VGPR j holds rows (j, j+8), N = lane%16.
// ---------------------------------------------------------------------------
__global__ void __launch_bounds__(256) gemm_f32_wmma(const float* __restrict__ A,
                                                     const float* __restrict__ Bm,
                                                     float* __restrict__ C,
                                                     int M, int N, int K) {
  const int wavesPerBlock = blockDim.x >> 5;
  const int tile = blockIdx.x * wavesPerBlock + (threadIdx.x >> 5);
  const int ntn = N >> 4;
  const int numTiles = (M >> 4) * ntn;
  if (tile >= numTiles) return;                 // wave-uniform exit (EXEC stays all-1s)
  const int m0 = (tile / ntn) << 4;
  const int n0 = (tile % ntn) << 4;
  const int lane = threadIdx.x & 31;
  const int half = lane >> 4;
  const int r = lane & 15;

  const float* Arow = A + (size_t)(m0 + r) * K + 2 * half;
  const float* Bcol = Bm + (size_t)(2 * half) * N + n0 + r;

  v8f acc = {0.f, 0.f, 0.f, 0.f, 0.f, 0.f, 0.f, 0.f};

#pragma unroll 4
  for (int kk = 0; kk < K; kk += 4) {
    v2f a, b;
    a[0] = Arow[kk];
    a[1] = Arow[kk + 1];
    b[0] = Bcol[(size_t)kk * N];
    b[1] = Bcol[(size_t)kk * N + N];
    acc = __builtin_amdgcn_wmma_f32_16x16x4_f32(
        false, a, false, b, (short)0, acc, false, false);
  }

  float* Crow = C + (size_t)n0 + r;
#pragma unroll
  for (int j = 0; j < 8; ++j)
    Crow[(size_t)(m0 + j + 8 * half) * N] = acc[j];
}

// ---------------------------------------------------------------------------
// Small projections: beta = 2*sigmoid(X@Wb), g = -exp(A_log)*softplus(X@Wa+dt_bias)
// ---------------------------------------------------------------------------
__global__ void __launch_bounds__(256) small_proj_kernel(
    const float* __restrict__ X, const float* __restrict__ Wb,
    const float* __restrict__ Wa, const float* __restrict__ A_log,
    const float* __restrict__ dt_bias, float* __restrict__ beta_out,
    float* __restrict__ g_out) {
  const int NCOL = NHn * Hh + Hh;               // 12
  int idx = blockIdx.x * blockDim.x + threadIdx.x;
  if (idx >= Mrows * NCOL) return;
  int row = idx / NCOL;
  int c = idx % NCOL;
  const float* xr = X + (size_t)row * HID;
  float dot = 0.f;
  if (c < NHn * Hh) {
    for (int k = 0; k < HID; ++k) dot += xr[k] * Wb[(size_t)k * (NHn * Hh) + c];
    beta_out[(size_t)row * (NHn * Hh) + c] = 2.f * sigmoidf_(dot);
  } else {
    int ca = c - NHn * Hh;
    for (int k = 0; k < HID; ++k) dot += xr[k] * Wa[(size_t)k * Hh + ca];
    float xv = dot + dt_bias[ca];
    float sp = (xv > 20.f) ? xv : log1pf(__expf(xv));
    g_out[(size_t)row * Hh + ca] = -__expf(A_log[ca]) * sp;
  }
}

// ---------------------------------------------------------------------------
// Causal depthwise conv1d (K=4) + SiLU. x,y: [B,T,C], w: [C,K]
// ---------------------------------------------------------------------------
__global__ void __launch_bounds__(256) conv_silu_kernel(const float* __restrict__ x,
                                                        const float* __restrict__ w,
                                                        float* __restrict__ y,
                                                        int C, int total) {
  int idx = blockIdx.x * blockDim.x + threadIdx.x;
  if (idx >= total) return;
  int c = idx % C;
  int bt = idx / C;
  int t = bt % Tt;
  float acc = 0.f;
#pragma unroll
  for (int i = 0; i < KSZ; ++i) {
    int tt = t - (KSZ - 1) + i;
    if (tt >= 0) acc += x[(size_t)(bt + (tt - t)) * C + c] * w[c * KSZ + i];
  }
  y[idx] = siluf_(acc);
}

// ---------------------------------------------------------------------------
// In-place l2 norm over groups of DK=128 contiguous floats; one wave32 per group.
// ---------------------------------------------------------------------------
__global__ void __launch_bounds__(256) l2norm_kernel(float* __restrict__ buf, int G) {
  int gid = (blockIdx.x * blockDim.x + threadIdx.x) >> 5;
  int lane = threadIdx.x & 31;
  if (gid >= G) return;
  size_t base = (size_t)gid * DK;
  float xv[4];
  float s = 0.f;
#pragma unroll
  for (int j = 0; j < 4; ++j) {
    xv[j] = buf[base + lane + 32 * j];
    s += xv[j] * xv[j];
  }
#pragma unroll
  for (int m = 16; m >= 1; m >>= 1) s += __shfl_xor(s, m, 32);   // wave32 reduce
  float sc = rsqrtf(s + 1e-6f);
#pragma unroll
  for (int j = 0; j < 4; ++j) buf[base + lane + 32 * j] = xv[j] * sc;
}

// ---------------------------------------------------------------------------
// Sequential gated-delta scan, DV-sharded.
// Grid: Bsz*Hh*NSLICE blocks of 256. Block owns state slice S[128 x SLC].
// Thread: col = slice*SLC + tid/8, part = tid&7, owns rows [part*16, part*16+16).
// k-reductions via __shfl_xor over the 8 parts (same wave, wave32).
// Global operands software-pipelined (prefetch t+1 into regs during compute of t).
// ---------------------------------------------------------------------------
__global__ void __launch_bounds__(256, 1) scan_kernel(
    const float* __restrict__ qc, const float* __restrict__ kc,
    const float* __restrict__ vc, const float* __restrict__ betap,
    const float* __restrict__ gdec, float* __restrict__ oscan) {
  const int slice = blockIdx.x % NSLICE;
  const int h = (blockIdx.x / NSLICE) % Hh;
  const int b = blockIdx.x / (NSLICE * Hh);
  const int tid = threadIdx.x;
  const int lcol = tid >> 3;            // 0..31 local column
  const int col = slice * SLC + lcol;   // global v column
  const int part = tid & (PARTS - 1);   // 0..7

  __shared__ __align__(16) float qs[DK];
  __shared__ __align__(16) float ks[NHn][DK];
  __shared__ float vs[NHn][SLC];
  __shared__ float betas[NHn];
  __shared__ float gs_s;

  float S[KPT];
#pragma unroll
  for (int k = 0; k < KPT; ++k) S[k] = 0.f;

  float rq = 0.f, rk = 0.f, rv = 0.f, rb = 0.f, rg = 0.f;
  auto prefetch = [&](int t) {
    const size_t bt = (size_t)b * Tt + t;
    if (tid < DK) rq = qc[(bt * Hh + h) * DK + tid];
    {
      int i = tid >> 7, j = tid & 127;
      rk = kc[((bt * NHn + i) * Hh + h) * DK + j];
    }
    if (tid < NHn * SLC) {
      int i = tid / SLC, j = tid % SLC;
      rv = vc[((bt * NHn + i) * Hh + h) * DV + slice * SLC + j];
    }
    if (tid < NHn) rb = betap[(bt * NHn + tid) * Hh + h];
    if (tid == 0) rg = gdec[bt * Hh + h];
  };

  prefetch(0);
  for (int t = 0; t < Tt; ++t) {
    // stage prefetched operands into LDS
    if (tid < DK) qs[tid] = rq;
    {
      int i = tid >> 7, j = tid & 127;
      ks[i][j] = rk;
    }
    if (tid < NHn * SLC) {
      int i = tid / SLC, j = tid % SLC;
      vs[i][j] = rv;
    }
    if (tid < NHn) betas[tid] = rb;
    if (tid == 0) gs_s = rg;
    __syncthreads();

    if (t + 1 < Tt) prefetch(t + 1);    // overlap next step's global loads

    const float decay = __expf(gs_s);
#pragma unroll
    for (int k = 0; k < KPT; ++k) S[k] *= decay;

#pragma unroll
    for (int i = 0; i < NHn; ++i) {
      const float4* k4 = reinterpret_cast<const float4*>(ks[i]);
      float4 kk[KPT / 4];
      float kv0 = 0.f, kv1 = 0.f, kv2 = 0.f, kv3 = 0.f;
#pragma unroll
      for (int j = 0; j < KPT / 4; ++j) {
        kk[j] = k4[part * (KPT / 4) + j];        // ds_load_b128 broadcast, cached
        kv0 += kk[j].x * S[4 * j + 0];
        kv1 += kk[j].y * S[4 * j + 1];
        kv2 += kk[j].z * S[4 * j + 2];
        kv3 += kk[j].w * S[4 * j + 3];
      }
      float kv = (kv0 + kv1) + (kv2 + kv3);
      kv += __shfl_xor(kv, 1, 32);
      kv += __shfl_xor(kv, 2, 32);
      kv += __shfl_xor(kv, 4, 32);               // all 8 parts now hold full kv
      const float u = betas[i] * (vs[i][lcol] - kv);
#pragma unroll
      for (int j = 0; j < KPT / 4; ++j) {
        S[4 * j + 0] += kk[j].x * u;
        S[4 * j + 1] += kk[j].y * u;
        S[4 * j + 2] += kk[j].z * u;
        S[4 * j + 3] += kk[j].w * u;
      }
    }

    const float4* q4 = reinterpret_cast<const float4*>(qs);
    float o0 = 0.f, o1 = 0.f, o2 = 0.f, o3 = 0.f;
#pragma unroll
    for (int j = 0; j < KPT / 4; ++j) {
      float4 qq = q4[part * (KPT / 4) + j];
      o0 += qq.x * S[4 * j + 0];
      o1 += qq.y * S[4 * j + 1];
      o2 += qq.z * S[4 * j + 2];
      o3 += qq.w * S[4 * j + 3];
    }
    float o = (o0 + o1) + (o2 + o3);
    o += __shfl_xor(o, 1, 32);
    o += __shfl_xor(o, 2, 32);
    o += __shfl_xor(o, 4, 32);
    if (part == 0)
      oscan[(((size_t)b * Tt + t) * Hh + h) * DV + col] = o;
    __syncthreads();                    // LDS reusable next iteration
  }
}

// ---------------------------------------------------------------------------
// Gated RMS-norm: one block of 256 per (b,t,h) group of DV=256.
// ---------------------------------------------------------------------------
__global__ void __launch_bounds__(256) norm_gate_kernel(
    const float* __restrict__ oscan, const float* __restrict__ gproj,
    const float* __restrict__ w, float* __restrict__ ogated) {
  __shared__ float red[256];
  const int tid = threadIdx.x;
  const size_t base = (size_t)blockIdx.x * DV;
  float val = oscan[base + tid];
  red[tid] = val * val;
  __syncthreads();
  for (int s = 128; s > 0; s >>= 1) {
    if (tid < s) red[tid] += red[tid + s];
    __syncthreads();
  }
  float rms = rsqrtf(red[0] / (float)DV + EPSc);
  float gate = gproj[base + tid];
  ogated[base + tid] = val * rms * w[tid] * siluf_(gate);
}

// ---------------------------------------------------------------------------
// host-side launch
// ---------------------------------------------------------------------------
extern "C" void kernel_launch(void* const* d_in, const int* in_sizes, int n_in,
                              void* d_out, int out_size, void* d_ws, size_t ws_size,
                              hipStream_t stream) {
  const float* X        = (const float*)d_in[0];
  const float* Wq       = (const float*)d_in[1];
  const float* Wk       = (const float*)d_in[2];
  const float* Wv       = (const float*)d_in[3];
  const float* Wb       = (const float*)d_in[4];
  const float* Wa       = (const float*)d_in[5];
  const float* Wg       = (const float*)d_in[6];
  const float* Wo       = (const float*)d_in[7];
  const float* A_log    = (const float*)d_in[8];
  const float* dt_bias  = (const float*)d_in[9];
  const float* wq_conv  = (const float*)d_in[10];
  const float* wk_conv  = (const float*)d_in[11];
  const float* wv_conv  = (const float*)d_in[12];
  const float* o_norm_w = (const float*)d_in[13];
  float* out = (float*)d_out;

  // workspace layout (floats)
  float* ws = (float*)d_ws;
  const size_t NQ = (size_t)Mrows * Hh * DK;          // 1 Mi
  const size_t NK = (size_t)Mrows * NHn * Hh * DK;    // 2 Mi
  const size_t NV = (size_t)Mrows * NHn * Hh * DV;    // 4 Mi
  const size_t NG = (size_t)Mrows * Hh * DV;          // 2 Mi
  float* qproj  = ws;
  float* kproj  = qproj + NQ;
  float* vproj  = kproj + NK;
  float* gproj  = vproj + NV;
  float* qc     = gproj + NG;
  float* kc     = qc + NQ;
  float* vc     = kc + NK;
  float* betab  = vc + NV;
  float* gdec   = betab + (size_t)Mrows * NHn * Hh;
  float* oscan  = qproj;               // reuse (projections dead after conv)
  float* ogated = kproj;               // reuse

  const int threads = 256;
  const int wavesPerBlock = threads / 32;

  auto tiles = [](int M, int N) { return (M / 16) * (N / 16); };
  auto gblocks = [&](int nt) { return (nt + wavesPerBlock - 1) / wavesPerBlock; };

  // 1) projection GEMMs (f32 WMMA)
  gemm_f32_wmma<<<gblocks(tiles(Mrows, Hh * DK)), threads, 0, stream>>>(
      X, Wq, qproj, Mrows, Hh * DK, HID);
  gemm_f32_wmma<<<gblocks(tiles(Mrows, NHn * Hh * DK)), threads, 0, stream>>>(
      X, Wk, kproj, Mrows, NHn * Hh * DK, HID);
  gemm_f32_wmma<<<gblocks(tiles(Mrows, NHn * Hh * DV)), threads, 0, stream>>>(
      X, Wv, vproj, Mrows, NHn * Hh * DV, HID);
  gemm_f32_wmma<<<gblocks(tiles(Mrows, Hh * DV)), threads, 0, stream>>>(
      X, Wg, gproj, Mrows, Hh * DV, HID);

  // 2) beta / g gates
  {
    int total = Mrows * (NHn * Hh + Hh);
    small_proj_kernel<<<(total + threads - 1) / threads, threads, 0, stream>>>(
        X, Wb, Wa, A_log, dt_bias, betab, gdec);
  }

  // 3) causal conv + SiLU
  {
    int Cq = Hh * DK, Ck = NHn * Hh * DK, Cv = NHn * Hh * DV;
    conv_silu_kernel<<<((int)NQ + threads - 1) / threads, threads, 0, stream>>>(
        qproj, wq_conv, qc, Cq, (int)NQ);
    conv_silu_kernel<<<((int)NK + threads - 1) / threads, threads, 0, stream>>>(
        kproj, wk_conv, kc, Ck, (int)NK);
    conv_silu_kernel<<<((int)NV + threads - 1) / threads, threads, 0, stream>>>(
        vproj, wv_conv, vc, Cv, (int)NV);
  }

  // 4) l2 norm of q, k over DK
  {
    int Gq = Mrows * Hh;
    int Gk = Mrows * NHn * Hh;
    l2norm_kernel<<<(Gq * 32 + threads - 1) / threads, threads, 0, stream>>>(qc, Gq);
    l2norm_kernel<<<(Gk * 32 + threads - 1) / threads, threads, 0, stream>>>(kc, Gk);
  }

  // 5) sequential gated-delta scan: DV-sharded, 64 blocks
  scan_kernel<<<Bsz * Hh * NSLICE, threads, 0, stream>>>(qc, kc, vc, betab, gdec, oscan);

  // 6) gated RMS norm
  norm_gate_kernel<<<Mrows * Hh, threads, 0, stream>>>(oscan, gproj, o_norm_w, ogated);

  // 7) output GEMM (f32 WMMA) into d_out
  gemm_f32_wmma<<<gblocks(tiles(Mrows, HID)), threads, 0, stream>>>(
      ogated, Wo, out, Mrows, HID, HID);
}